// Loss_70695161692240
// MI455X (gfx1250) — compile-verified
//
#include <hip/hip_runtime.h>
#include <math.h>

#define NQ 7
#define Dm 128
#define Bsz 1024
#define Tn 8256            // D*(D+1)/2
#define LD 129             // +1 pad: column walks hit distinct LDS banks
#define NTHREADS 256
#define SWEEPS 10
#define DS_C 0.25f

typedef float v2f __attribute__((ext_vector_type(2)));
typedef float v8f __attribute__((ext_vector_type(8)));

// GATES: 0 = H, 1 = H@S^dagger, 2 = I   (s = 1/sqrt(2))
__constant__ float GR[3][2][2] = {
  {{0.70710678f, 0.70710678f}, {0.70710678f, -0.70710678f}},
  {{0.70710678f, 0.f},         {0.70710678f,  0.f}},
  {{1.f, 0.f},                 {0.f, 1.f}}};
__constant__ float GI[3][2][2] = {
  {{0.f, 0.f},          {0.f, 0.f}},
  {{0.f, -0.70710678f}, {0.f, 0.70710678f}},
  {{0.f, 0.f},          {0.f, 0.f}}};

// Round-robin tournament pairing: 127 rounds x 64 disjoint pairs covers all C(128,2)
__device__ __forceinline__ void pair_pq(int r, int j, int& p, int& q) {
  if (j == 0) { p = 127; q = r % 127; }
  else        { p = (r + j) % 127; q = (r + 127 - j) % 127; }
}

// C(128x128) = op(A) * op(B); LDS-resident fp32 via V_WMMA_F32_16X16X4_F32.
// Register-blocked: one A fragment feeds 4 independent N-tile accumulators
// (10 ds_loads per 4 wmma instead of 16), so the 4 acc chains pipeline.
// A-frag 16x4: lanes 0-15 K={0,1}, lanes 16-31 K={2,3}; B-frag mirrored, N over lanes.
// C/D 8 VGPRs: lanes 0-15 M=r, lanes 16-31 M=8+r.
__device__ __forceinline__ void mm128(const float* A, const float* Bm, float* Cm,
                                      bool tA, bool tB, int tid) {
  const int wave = tid >> 5;
  const int lane = tid & 31;
  const int half = lane >> 4;
  const int lm   = lane & 15;
  const int tileM = wave * 16;               // 8 waves -> 8 row strips
  const int m = tileM + lm;
  for (int tg = 0; tg < 2; ++tg) {           // two groups of 4 N-tiles
    v8f acc[4];
#pragma unroll
    for (int t = 0; t < 4; ++t) acc[t] = (v8f){0.f,0.f,0.f,0.f,0.f,0.f,0.f,0.f};
    for (int kk = 0; kk < Dm; kk += 4) {
      const int k0 = kk + half * 2;
      v2f a;
      a.x = tA ? A[(k0    ) * LD + m] : A[m * LD + (k0    )];
      a.y = tA ? A[(k0 + 1) * LD + m] : A[m * LD + (k0 + 1)];
#pragma unroll
      for (int t = 0; t < 4; ++t) {
        const int n = tg * 64 + t * 16 + lm;
        v2f bf;
        bf.x = tB ? Bm[n * LD + (k0    )] : Bm[(k0    ) * LD + n];
        bf.y = tB ? Bm[n * LD + (k0 + 1)] : Bm[(k0 + 1) * LD + n];
        acc[t] = __builtin_amdgcn_wmma_f32_16x16x4_f32(false, a, false, bf,
                                                       (short)0, acc[t], false, false);
      }
    }
#pragma unroll
    for (int t = 0; t < 4; ++t)
#pragma unroll
      for (int r = 0; r < 8; ++r)
        Cm[(tileM + half * 8 + r) * LD + (tg * 64 + t * 16 + lm)] = acc[t][r];
  }
}

__global__ void zero_out_kernel(float* out) {
  if (threadIdx.x == 0) out[0] = 0.f;
}

__global__ __launch_bounds__(NTHREADS)
void qloss_kernel(const float* __restrict__ flat, const float* __restrict__ timev,
                  const float* __restrict__ state, const int* __restrict__ tgt,
                  const int* __restrict__ basis, float* __restrict__ out) {
  __shared__ float b0[Dm * LD];   // A -> matmul scratch
  __shared__ float b1[Dm * LD];   // V (eigenvectors)
  __shared__ float b2[Dm * LD];   // rho_r -> rho~_r -> M_r -> ev_r
  __shared__ float b3[Dm * LD];   // rho_i -> rho~_i -> M_i -> ev_i
  __shared__ float ca[Dm], sa[Dm];
  __shared__ float2 cs[64];       // per-round rotation (c,s)
  __shared__ int2   pq[64];       // per-round pair table (no per-element %127)
  __shared__ float prob[Dm];

  const int b   = blockIdx.x;
  const int tid = threadIdx.x;
  const float* fb = flat + (size_t)b * Tn;

  // ---- Build symmetric A from tril-flat; V = I ----
  for (int idx = tid; idx < Dm * Dm; idx += NTHREADS) {
    int r = idx >> 7, c = idx & 127;
    int hi = r >= c ? r : c, lo = r >= c ? c : r;
    b0[r * LD + c] = fb[(hi * (hi + 1)) / 2 + lo];
    b1[r * LD + c] = (r == c) ? 1.f : 0.f;
  }
  // ---- Prefetch rho into LDS now (b2/b3 idle during eigh): the ~128KB of HBM
  //      traffic per block fully hides behind the Jacobi compute. ----
  const float2* st2 = (const float2*)(state + (size_t)b * Dm * Dm * 2);
  for (int idx = tid; idx < Dm * Dm; idx += NTHREADS) {
    float2 v = st2[idx];
    int r = idx >> 7, c = idx & 127;
    b2[r * LD + c] = v.x;
    b3[r * LD + c] = v.y;
  }
  __syncthreads();

  // ---- Parallel cyclic Jacobi eigensolver (fp32, LDS) ----
  const int kcol  = tid & 127;      // loop-invariant column/row index per thread
  const int jbase = tid >> 7;       // 0 or 1: pair parity handled by this thread
  for (int sw = 0; sw < SWEEPS; ++sw) {
    for (int rd = 0; rd < 127; ++rd) {
      if (tid < 64) {
        int p, q; pair_pq(rd, tid, p, q);
        float app = b0[p * LD + p], aqq = b0[q * LD + q], apq = b0[p * LD + q];
        float c = 1.f, s = 0.f;
        if (fabsf(apq) > 1e-20f) {
          float th = (aqq - app) / (2.f * apq);
          float t  = copysignf(1.f, th) / (fabsf(th) + sqrtf(1.f + th * th));
          c = 1.f / sqrtf(1.f + t * t);
          s = t * c;
        }
        cs[tid] = make_float2(c, s);
        pq[tid] = make_int2(p, q);
      }
      __syncthreads();
      // G^T * A : row pairs (rows disjoint across the 64 pairs)
      for (int j = jbase; j < 64; j += 2) {
        int2   P = pq[j];
        float2 C = cs[j];
        float ap = b0[P.x * LD + kcol], aq = b0[P.y * LD + kcol];
        b0[P.x * LD + kcol] = C.x * ap - C.y * aq;
        b0[P.y * LD + kcol] = C.y * ap + C.x * aq;
      }
      __syncthreads();
      // (G^T A) * G and V * G : column pairs (cols disjoint)
      for (int j = jbase; j < 64; j += 2) {
        int2   P = pq[j];
        float2 C = cs[j];
        float ap = b0[kcol * LD + P.x], aq = b0[kcol * LD + P.y];
        b0[kcol * LD + P.x] = C.x * ap - C.y * aq;
        b0[kcol * LD + P.y] = C.y * ap + C.x * aq;
        float vp = b1[kcol * LD + P.x], vq = b1[kcol * LD + P.y];
        b1[kcol * LD + P.x] = C.x * vp - C.y * vq;
        b1[kcol * LD + P.y] = C.y * vp + C.x * vq;
      }
      __syncthreads();
    }
  }

  // ---- Angles from eigenvalues ----
  if (tid < Dm) {
    float a = DS_C * timev[b] * b0[tid * LD + tid];
    ca[tid] = cosf(a);
    sa[tid] = sinf(a);
  }
  __syncthreads();

  // ---- rho~ = V^T rho V  (4 fp32 WMMA matmuls) ----
  mm128(b1, b2, b0, true,  false, tid);  __syncthreads(); // T  = V^T * rho_r
  mm128(b0, b1, b2, false, false, tid);  __syncthreads(); // rho~_r = T * V
  mm128(b1, b3, b0, true,  false, tid);  __syncthreads(); // T2 = V^T * rho_i
  mm128(b0, b1, b3, false, false, tid);  __syncthreads(); // rho~_i = T2 * V

  // ---- elementwise phase twist: X_{kl} *= e^{-i(a_k - a_l)} ----
  for (int idx = tid; idx < Dm * Dm; idx += NTHREADS) {
    int k = idx >> 7, l = idx & 127;
    float c = ca[k] * ca[l] + sa[k] * sa[l];   // cos(a_k - a_l)
    float s = sa[k] * ca[l] - ca[k] * sa[l];   // sin(a_k - a_l)
    float xr = b2[k * LD + l], xi = b3[k * LD + l];
    b2[k * LD + l] = xr * c + xi * s;
    b3[k * LD + l] = xi * c - xr * s;
  }
  __syncthreads();

  // ---- ev = V M V^T  (4 fp32 WMMA matmuls) ----
  mm128(b1, b2, b0, false, false, tid);  __syncthreads(); // T3 = V * M_r
  mm128(b0, b1, b2, false, true,  tid);  __syncthreads(); // ev_r = T3 * V^T
  mm128(b1, b3, b0, false, false, tid);  __syncthreads(); // T4 = V * M_i
  mm128(b0, b1, b3, false, true,  tid);  __syncthreads(); // ev_i = T4 * V^T

  // ---- F ev F^H via 7 per-qubit 2x2 butterflies per side (qubit q <-> bit 6-q) ----
  const int* bb = basis + b * NQ;
  for (int qq = 0; qq < NQ; ++qq) {
    int g = bb[qq];
    float g00r = GR[g][0][0], g01r = GR[g][0][1], g10r = GR[g][1][0], g11r = GR[g][1][1];
    float g00i = GI[g][0][0], g01i = GI[g][0][1], g10i = GI[g][1][0], g11i = GI[g][1][1];
    int mask = 1 << (6 - qq);
    // left multiply: row pairs
    for (int pr = jbase; pr < 64; pr += 2) {
      int i0 = ((pr & ~(mask - 1)) << 1) | (pr & (mask - 1));
      int i1 = i0 | mask;
      float x0r = b2[i0 * LD + kcol], x0i = b3[i0 * LD + kcol];
      float x1r = b2[i1 * LD + kcol], x1i = b3[i1 * LD + kcol];
      b2[i0 * LD + kcol] = g00r * x0r - g00i * x0i + g01r * x1r - g01i * x1i;
      b3[i0 * LD + kcol] = g00r * x0i + g00i * x0r + g01r * x1i + g01i * x1r;
      b2[i1 * LD + kcol] = g10r * x0r - g10i * x0i + g11r * x1r - g11i * x1i;
      b3[i1 * LD + kcol] = g10r * x0i + g10i * x0r + g11r * x1i + g11i * x1r;
    }
    __syncthreads();
    // right multiply by G^H: column pairs, conjugated gate
    for (int pc = jbase; pc < 64; pc += 2) {
      int j0 = ((pc & ~(mask - 1)) << 1) | (pc & (mask - 1));
      int j1 = j0 | mask;
      float y0r = b2[kcol * LD + j0], y0i = b3[kcol * LD + j0];
      float y1r = b2[kcol * LD + j1], y1i = b3[kcol * LD + j1];
      b2[kcol * LD + j0] = y0r * g00r + y0i * g00i + y1r * g01r + y1i * g01i;
      b3[kcol * LD + j0] = y0i * g00r - y0r * g00i + y1i * g01r - y1r * g01i;
      b2[kcol * LD + j1] = y0r * g10r + y0i * g10i + y1r * g11r + y1i * g11i;
      b3[kcol * LD + j1] = y0i * g10r - y0r * g10i + y1i * g11r - y1r * g11i;
    }
    __syncthreads();
  }

  // ---- diag -> probs -> target log-prob -> atomic accumulate ----
  if (tid < Dm) {
    float dr = b2[tid * LD + tid], di = b3[tid * LD + tid];
    prob[tid] = dr * dr + di * di;
  }
  __syncthreads();
  if (tid == 0) {
    float sum = 0.f;
    for (int i = 0; i < Dm; ++i) sum += prob[i];
    float tp = prob[tgt[b]] / sum;
    tp = fmaxf(tp, 1e-12f);
    atomicAdd(out, -logf(tp) * (1.f / ((float)Bsz * (float)NQ)));
  }
}

extern "C" void kernel_launch(void* const* d_in, const int* in_sizes, int n_in,
                              void* d_out, int out_size, void* d_ws, size_t ws_size,
                              hipStream_t stream) {
  (void)in_sizes; (void)n_in; (void)out_size; (void)d_ws; (void)ws_size;
  const float* flat  = (const float*)d_in[0];
  const float* timev = (const float*)d_in[1];
  const float* state = (const float*)d_in[2];
  const int*   tgt   = (const int*)d_in[3];
  const int*   basis = (const int*)d_in[4];
  float* out = (float*)d_out;

  zero_out_kernel<<<1, 32, 0, stream>>>(out);
  qloss_kernel<<<Bsz, NTHREADS, 0, stream>>>(flat, timev, state, tgt, basis, out);
}